// Model_1537598292268
// MI455X (gfx1250) — compile-verified
//
#include <hip/hip_runtime.h>
#include <math.h>

typedef __attribute__((ext_vector_type(2))) float v2f;
typedef __attribute__((ext_vector_type(8))) float v8f;

#define NT 256
#define SM_TOTAL 44032  // floats -> 176128 bytes dynamic LDS (< 320KB/WGP)

// float-index offsets inside dynamic LDS
#define XS_OFF   0        // x slice: 23 rows x 1024 (row 22 = zero pad, col shift +1)
#define V2_OFF   0        // phase>=3: reuses XS region, 16 x 1024
#define V1_OFF   23552    // 16 x 1088 (data at cols 32..1031, zero halo for conv)
#define SW_OFF   40960    // 16 x 24 spatial depthwise weights
#define WA_OFF   41344    // 16 x 8 stem weights (taps j<6)
#define STG_OFF  41472
#define STB_OFF  41488
#define G1_OFF   41504
#define B1_OFF   41520
#define G2_OFF   41536
#define B2_OFF   41552
#define ST1_OFF  41568    // 32: sum / sumsq
#define ST2_OFF  41600    // 32
#define MU1_OFF  41632
#define C1_OFF   41648
#define MU2_OFF  41664
#define C2_OFF   41680
#define TW_OFF   41696    // 16 x 64 temporal depthwise kernels
#define PB_OFF   42720    // 16 x 40 pooled branch output
#define Z_OFF    43360    // 5 x 40
#define MU3_OFF  43560
#define C3_OFF   43568
#define Z1_OFF   43576    // 5 x 36
#define Z2_OFF   43756    // 5 x 27
#define P2_OFF   43896    // 125

__device__ __forceinline__ float eluf(float x) {
  // branchless: single v_exp_f32 (TRANS, co-executes) + cndmask
  float e = __expf(x) - 1.0f;
  return x > 0.0f ? x : e;
}
__device__ __forceinline__ float sigm(float x) {
  return __builtin_amdgcn_rcpf(1.0f + __expf(-x));
}
__device__ __forceinline__ float simam_g(float v, float mu, float cinv) {
  float d = v - mu;
  float e = d * d * cinv + 1e-4f;          // lam = 1e-4
  return sigm(__builtin_amdgcn_rcpf(e + 1e-12f)) * v;
}

extern "C" __global__ __launch_bounds__(NT)
void eeg_fused(const float* __restrict__ x,
  const float* __restrict__ stw, const float* __restrict__ stg, const float* __restrict__ stb,
  const float* __restrict__ b1sw, const float* __restrict__ b1g1, const float* __restrict__ b1b1,
  const float* __restrict__ b1tw, const float* __restrict__ b1g2, const float* __restrict__ b1b2,
  const float* __restrict__ b2sw, const float* __restrict__ b2g1, const float* __restrict__ b2b1,
  const float* __restrict__ b2tw, const float* __restrict__ b2g2, const float* __restrict__ b2b2,
  const float* __restrict__ b3sw, const float* __restrict__ b3g1, const float* __restrict__ b3b1,
  const float* __restrict__ b3tw, const float* __restrict__ b3g2, const float* __restrict__ b3b2,
  const float* __restrict__ pww, const float* __restrict__ pwg, const float* __restrict__ pwb,
  const float* __restrict__ d1w, const float* __restrict__ d1ow, const float* __restrict__ d1ob,
  const float* __restrict__ d2w, const float* __restrict__ d2ow, const float* __restrict__ d2ob,
  const float* __restrict__ fcw, const float* __restrict__ fcb,
  float* __restrict__ out)
{
  extern __shared__ float sm[];
  const int tid  = threadIdx.x;
  const int b    = blockIdx.x;
  const int lane = tid & 31;
  const int w    = tid >> 5;
  const int half = lane >> 4;
  const int ln   = lane & 15;
  const float rs = rsqrtf(1.0f + 1e-5f);   // BN eval scale 1/sqrt(var(=1)+eps)

  // ---------------- setup: zero LDS ----------------
  for (int i = tid; i < SM_TOTAL; i += NT) sm[i] = 0.0f;
  __syncthreads();

  // stage x[b] (22 x 1000), column shift +1 => zero padding at t=-1 / t>=1000; row 22 stays zero
  const float* xb = x + (size_t)b * 22 * 1000;
  for (int i = tid; i < 22 * 1000; i += NT) {
    if (i + 4 * NT < 22 * 1000) __builtin_prefetch(xb + i + 4 * NT, 0, 0);
    int ch = i / 1000, t = i - ch * 1000;
    sm[XS_OFF + ch * 1024 + t + 1] = xb[i];
  }
  // stem weights -> WA[16][8] (pad rows/taps with zeros)
  for (int i = tid; i < 90; i += NT) {
    int oc = i / 6, j = i - oc * 6;
    sm[WA_OFF + oc * 8 + j] = stw[i];
  }
  for (int i = tid; i < 15; i += NT) {
    sm[STG_OFF + i] = stg[i] * rs;
    sm[STB_OFF + i] = stb[i];
  }
  // spatial depthwise weights: k=0..14 maps to branch k/5, channel k%5
  for (int i = tid; i < 330; i += NT) {
    int k = i / 22, h = i - k * 22;
    const float* p = (k < 5) ? b1sw : (k < 10 ? b2sw : b3sw);
    sm[SW_OFF + k * 24 + h] = p[(k % 5) * 22 + h];
  }
  for (int i = tid; i < 15; i += NT) {
    int c = i % 5;
    const float* g1  = (i < 5) ? b1g1 : (i < 10 ? b2g1 : b3g1);
    const float* bb1 = (i < 5) ? b1b1 : (i < 10 ? b2b1 : b3b1);
    const float* g2  = (i < 5) ? b1g2 : (i < 10 ? b2g2 : b3g2);
    const float* bb2 = (i < 5) ? b1b2 : (i < 10 ? b2b2 : b3b2);
    sm[G1_OFF + i] = g1[c] * rs;  sm[B1_OFF + i] = bb1[c];
    sm[G2_OFF + i] = g2[c] * rs;  sm[B2_OFF + i] = bb2[c];
  }
  // temporal depthwise kernels (zero padded to 64 taps)
  for (int i = tid; i < 15 * 64; i += NT) {
    int k = i / 64, j = i - k * 64;
    int kt = (k < 5) ? 15 : (k < 10 ? 31 : 63);
    const float* p = (k < 5) ? b1tw : (k < 10 ? b2tw : b3tw);
    sm[TW_OFF + k * 64 + j] = (j < kt) ? p[(k % 5) * kt + j] : 0.0f;
  }
  __syncthreads();

  // ---------------- phase 1: stem GEMM via WMMA + BN/ELU + spatial depthwise ----------------
  // A (16x4 f32): lane = M, dword0 holds K=2*half, dword1 holds K=2*half+1 (per ISA layout)
  v2f a0, a1;
  a0.x = sm[WA_OFF + ln * 8 + 2 * half];
  a0.y = sm[WA_OFF + ln * 8 + 2 * half + 1];
  a1.x = half ? 0.0f : sm[WA_OFF + ln * 8 + 4];
  a1.y = half ? 0.0f : sm[WA_OFF + ln * 8 + 5];

  float sgc[8], sbc[8], g1c[8], b1c[8];
#pragma unroll
  for (int r = 0; r < 8; ++r) {
    int k = r + 8 * half;            // C/D layout: vgpr r, lane-half -> M = r + 8*half
    sgc[r] = sm[STG_OFF + k];
    sbc[r] = sm[STB_OFF + k];
    g1c[r] = sm[G1_OFF + k];
    b1c[r] = sm[B1_OFF + k];
  }
  float sumv[8], sumv2[8];
#pragma unroll
  for (int r = 0; r < 8; ++r) { sumv[r] = 0.0f; sumv2[r] = 0.0f; }

  // guaranteed-zero LDS cells for upper-half lanes of the K={4,5} chunk:
  // V1 halo cols 0..31 are never written during phase 1/2.
  const int zbase = V1_OFF - 1024;   // zbase+1024 / zbase+1025 -> V1 cols 0,1 (zero)

  for (int tile = w; tile < 63; tile += 8) {   // 63 tiles of 16 t cover T=1000 (+pad)
    const int tcol = tile * 16 + ln;           // this lane's N -> time index
    float sacc[8];
#pragma unroll
    for (int r = 0; r < 8; ++r) sacc[r] = 0.0f;

    for (int ch = 0; ch < 22; ++ch) {
      const int base  = XS_OFF + ch * 1024 + tcol + 1;  // x[ch][tcol] (shifted)
      const int base1 = half ? zbase : base;            // addr-select, no EXEC divergence
      // B (4x16 f32): lane = N, dword0 K=2*half, dword1 K=2*half+1; taps j==K
      v2f bx0, bx1;
      bx0.x = sm[base + (half ? 1 : -1)];     // j0: x[ch][t-1]   | j2: x[ch][t+1]
      bx0.y = sm[base + (half ? 1023 : 0)];   // j1: x[ch][t]     | j3: x[ch+1][t-1]
      bx1.x = sm[base1 + 1024];               // j4: x[ch+1][t]   | zero
      bx1.y = sm[base1 + 1025];               // j5: x[ch+1][t+1] | zero

      v8f acc = {0.0f,0.0f,0.0f,0.0f,0.0f,0.0f,0.0f,0.0f};
      acc = __builtin_amdgcn_wmma_f32_16x16x4_f32(false, a0, false, bx0, (short)0, acc, false, false);
      acc = __builtin_amdgcn_wmma_f32_16x16x4_f32(false, a1, false, bx1, (short)0, acc, false, false);

#pragma unroll
      for (int r = 0; r < 8; ++r) {
        float av = eluf(acc[r] * sgc[r] + sbc[r]);              // stem BN + ELU
        sacc[r] += av * sm[SW_OFF + (r + 8 * half) * 24 + ch];  // spatial depthwise
      }
    }
    const int valid = (tcol < 1000);
#pragma unroll
    for (int r = 0; r < 8; ++r) {
      float v  = eluf(sacc[r] * g1c[r] + b1c[r]);               // branch BN1 + ELU
      float vm = valid ? v : 0.0f;
      sumv[r]  += vm;
      sumv2[r] += vm * vm;
      if (valid) sm[V1_OFF + (r + 8 * half) * 1088 + 32 + tcol] = v;
    }
  }
  // reduce SimAM stats within each 16-lane half (same k), then combine across waves
#pragma unroll
  for (int r = 0; r < 8; ++r) {
#pragma unroll
    for (int m = 1; m < 16; m <<= 1) {
      sumv[r]  += __shfl_xor(sumv[r],  m, 32);
      sumv2[r] += __shfl_xor(sumv2[r], m, 32);
    }
  }
  if (ln == 0) {
#pragma unroll
    for (int r = 0; r < 8; ++r) {
      int k = r + 8 * half;
      atomicAdd(&sm[ST1_OFF + k],      sumv[r]);
      atomicAdd(&sm[ST1_OFF + 16 + k], sumv2[r]);
    }
  }
  __syncthreads();

  if (tid < 15) {
    float mu  = sm[ST1_OFF + tid] * (1.0f / 1000.0f);
    float var = fmaxf(sm[ST1_OFF + 16 + tid] * (1.0f / 1000.0f) - mu * mu, 0.0f);
    sm[MU1_OFF + tid] = mu;
    sm[C1_OFF + tid]  = 1.0f / (2.0f * var + 1e-12f);
  }
  __syncthreads();

  // SimAM #1 in place
  for (int i = tid; i < 15 * 1000; i += NT) {
    int k = i / 1000, t = i - k * 1000;
    int a = V1_OFF + k * 1088 + 32 + t;
    sm[a] = simam_g(sm[a], sm[MU1_OFF + k], sm[C1_OFF + k]);
  }
  __syncthreads();

  // ---------------- phase 3: temporal depthwise conv + BN2/ELU (one wave per channel) ----------------
  for (int k = w; k < 15; k += 8) {
    int kt = (k < 5) ? 15 : (k < 10 ? 31 : 63);
    int hw = kt >> 1;
    float g2c = sm[G2_OFF + k], b2c = sm[B2_OFF + k];
    float su = 0.0f, sq = 0.0f;
    for (int t = lane; t < 1000; t += 32) {
      float u = 0.0f;
      int rb = V1_OFF + k * 1088 + 32 + t - hw;   // zero halo handles 'same' padding
      for (int j = 0; j < kt; ++j)
        u += sm[rb + j] * sm[TW_OFF + k * 64 + j];
      float v = eluf(u * g2c + b2c);
      su += v; sq += v * v;
      sm[V2_OFF + k * 1024 + t] = v;              // reuses dead XS region
    }
#pragma unroll
    for (int m = 1; m < 32; m <<= 1) {
      su += __shfl_xor(su, m, 32);
      sq += __shfl_xor(sq, m, 32);
    }
    if (lane == 0) { sm[ST2_OFF + k] = su; sm[ST2_OFF + 16 + k] = sq; }
  }
  __syncthreads();

  if (tid < 15) {
    float mu  = sm[ST2_OFF + tid] * (1.0f / 1000.0f);
    float var = fmaxf(sm[ST2_OFF + 16 + tid] * (1.0f / 1000.0f) - mu * mu, 0.0f);
    sm[MU2_OFF + tid] = mu;
    sm[C2_OFF + tid]  = 1.0f / (2.0f * var + 1e-12f);
  }
  __syncthreads();

  // ---------------- phase 4: SimAM #2 + avgpool 1000 -> 40 ----------------
  for (int i = tid; i < 15 * 40; i += NT) {
    int k = i / 40, p = i - k * 40;
    float mu = sm[MU2_OFF + k], ci = sm[C2_OFF + k];
    float s = 0.0f;
    int base = V2_OFF + k * 1024 + p * 25;
    for (int j = 0; j < 25; ++j) s += simam_g(sm[base + j], mu, ci);
    sm[PB_OFF + k * 40 + p] = s * (1.0f / 25.0f);
  }
  __syncthreads();

  // ---------------- phase 5: head ----------------
  // pointwise 15->5 + BN/ELU
  for (int i = tid; i < 200; i += NT) {
    int o = i / 40, t = i - o * 40;
    float q = 0.0f;
    for (int c = 0; c < 15; ++c) q += pww[o * 15 + c] * sm[PB_OFF + c * 40 + t];
    sm[Z_OFF + i] = eluf(q * (pwg[o] * rs) + pwb[o]);
  }
  __syncthreads();
  if (tid < 5) {
    float s = 0.0f, sq = 0.0f;
    for (int t = 0; t < 40; ++t) { float v = sm[Z_OFF + tid * 40 + t]; s += v; sq += v * v; }
    float mu  = s * (1.0f / 40.0f);
    float var = fmaxf(sq * (1.0f / 40.0f) - mu * mu, 0.0f);
    sm[MU3_OFF + tid] = mu;
    sm[C3_OFF + tid]  = 1.0f / (2.0f * var + 1e-12f);
  }
  __syncthreads();
  for (int i = tid; i < 200; i += NT) {
    int o = i / 40;
    sm[Z_OFF + i] = simam_g(sm[Z_OFF + i], sm[MU3_OFF + o], sm[C3_OFF + o]);
  }
  __syncthreads();

  // deformable depthwise conv 1: (5,40) -> (5,36), K=5
  for (int i = tid; i < 5 * 36; i += NT) {
    int c = i / 36, t = i - c * 36;
    float acc = 0.0f;
    for (int k = 0; k < 5; ++k) {
      float off = d1ob[c * 5 + k];
      for (int j = 0; j < 5; ++j)
        off += sm[Z_OFF + c * 40 + t + j] * d1ow[(c * 5 + k) * 5 + j];
      float pos = (float)(t + k) + off;
      float fl = floorf(pos);
      float f  = pos - fl;
      int  i0  = (int)fl;
      float gl = (i0 >= 0     && i0     < 40) ? sm[Z_OFF + c * 40 + i0]     : 0.0f;
      float gh = (i0 + 1 >= 0 && i0 + 1 < 40) ? sm[Z_OFF + c * 40 + i0 + 1] : 0.0f;
      acc += (gl * (1.0f - f) + gh * f) * d1w[c * 5 + k];
    }
    sm[Z1_OFF + i] = acc;
  }
  __syncthreads();

  // deformable depthwise conv 2: (5,36) -> (5,27), K=10
  for (int i = tid; i < 5 * 27; i += NT) {
    int c = i / 27, t = i - c * 27;
    float acc = 0.0f;
    for (int k = 0; k < 10; ++k) {
      float off = d2ob[c * 10 + k];
      for (int j = 0; j < 10; ++j)
        off += sm[Z1_OFF + c * 36 + t + j] * d2ow[(c * 10 + k) * 10 + j];
      float pos = (float)(t + k) + off;
      float fl = floorf(pos);
      float f  = pos - fl;
      int  i0  = (int)fl;
      float gl = (i0 >= 0     && i0     < 36) ? sm[Z1_OFF + c * 36 + i0]     : 0.0f;
      float gh = (i0 + 1 >= 0 && i0 + 1 < 36) ? sm[Z1_OFF + c * 36 + i0 + 1] : 0.0f;
      acc += (gl * (1.0f - f) + gh * f) * d2w[c * 10 + k];
    }
    sm[Z2_OFF + i] = acc;
  }
  __syncthreads();

  // adaptive pool 27 -> 25 (torch windowing)
  for (int i = tid; i < 125; i += NT) {
    int c = i / 25, p = i - c * 25;
    int s = (p * 27) / 25;
    int e = ((p + 1) * 27 + 24) / 25;
    float acc = 0.0f;
    for (int j = s; j < e; ++j) acc += sm[Z2_OFF + c * 27 + j];
    sm[P2_OFF + i] = acc / (float)(e - s);
  }
  __syncthreads();

  // fully connected: (125) -> (4)
  if (tid < 4) {
    float acc = fcb[tid];
    for (int m = 0; m < 125; ++m) acc += sm[P2_OFF + m] * fcw[tid * 125 + m];
    out[b * 4 + tid] = acc;
  }
}

extern "C" void kernel_launch(void* const* d_in, const int* in_sizes, int n_in,
                              void* d_out, int out_size, void* d_ws, size_t ws_size,
                              hipStream_t stream) {
  (void)n_in; (void)out_size; (void)d_ws; (void)ws_size;
  const float* p[33];
  for (int i = 0; i < 33; ++i) p[i] = (const float*)d_in[i];
  const int B = in_sizes[0] / (22 * 1000);
  const size_t shmem = (size_t)SM_TOTAL * sizeof(float);
  hipFuncSetAttribute(reinterpret_cast<const void*>(eeg_fused),
                      hipFuncAttributeMaxDynamicSharedMemorySize, (int)shmem);
  eeg_fused<<<B, NT, shmem, stream>>>(
      p[0],  p[1],  p[2],  p[3],
      p[4],  p[5],  p[6],  p[7],  p[8],  p[9],
      p[10], p[11], p[12], p[13], p[14], p[15],
      p[16], p[17], p[18], p[19], p[20], p[21],
      p[22], p[23], p[24],
      p[25], p[26], p[27],
      p[28], p[29], p[30],
      p[31], p[32],
      (float*)d_out);
}